// OnlineAverager_62680752718461
// MI455X (gfx1250) — compile-verified
//
#include <hip/hip_runtime.h>

// ---- problem constants (match reference) ------------------------------
#define USZ     4096                    // UPDATE_SIZE
#define BATCH   256
#define NUPD    8
#define NCH     2
#define KEEP    (NUPD * USZ)            // 32768
#define NU      (BATCH + NUPD - 1)      // 263 u-blocks in SNAP
#define SNAP    (NU * USZ)              // 1077248
#define OUT_LEN (BATCH * USZ)           // 1048576
#define REST    ((NUPD - 1) * USZ)      // 28672

typedef __attribute__((ext_vector_type(4))) float f32x4;

// Async global->LDS DMA of one float4 per lane (gfx1250 ASYNCcnt path).
__device__ __forceinline__ void async_ld_b128(unsigned ldsOff, const float* gp)
{
    asm volatile("global_load_async_to_lds_b128 %0, %1, off"
                 :: "v"(ldsOff), "v"((unsigned long long)gp) : "memory");
}

// acc[c, u*4096+r] = snapshot[c, u*4096+r] + 0.125 * sum_k update[u-k, c, k*4096+r]
// grid = (4 quarter-row tiles, 263 u-rows, 2 channels); block = 256 thr (8 wave32).
// Interior rows (u = 7..255) use a fully unrolled 8-deep async DMA burst.
__global__ __launch_bounds__(256) void oa_main(const float* __restrict__ update,
                                               const float* __restrict__ snapshot,
                                               float* __restrict__ out)
{
    __shared__ float sbuf[NUPD * 1024];            // 8 segments x 4 KB = 32 KB

    const int t = threadIdx.x;                     // 0..255
    const int u = blockIdx.y;                      // 0..262  (no div/mod needed)
    const int c = blockIdx.z;                      // 0..1
    const int r = blockIdx.x * 1024 + t * 4;       // float offset within u-row

    // LDS byte offset of this lane's slot: flat LDS pointer low 32 bits are the
    // wave-relative LDS address (ISA 10.2).
    const unsigned lds0 = (unsigned)(unsigned long long)(const void*)&sbuf[t * 4];

    // Per-block-uniform k range: k in [max(0,u-255), min(7,u)].
    const int k_lo = (u > BATCH - 1) ? (u - (BATCH - 1)) : 0;
    const int k_hi = (u < NUPD - 1) ? u : (NUPD - 1);

    const float* gbase = update + ((size_t)(u * NCH + c) * KEEP + (size_t)r);
    // k -> k+1 moves one row back in b and one segment forward in j:
    const long long kstep = -(long long)(NCH * KEEP) + (long long)USZ;  // -61440 floats

    f32x4 sum = (f32x4)0.f;

    if (k_lo == 0 && k_hi == NUPD - 1) {
        // ---- interior fast path: straight-line 8-deep async burst ----
#pragma unroll
        for (int k = 0; k < NUPD; ++k)
            async_ld_b128(lds0 + (unsigned)(k * 4096), gbase + (long long)k * kstep);

        // snapshot term streams once -> NT load overlaps the DMA
        f32x4 acc = __builtin_nontemporal_load(
            (const f32x4*)(snapshot + (size_t)c * SNAP + (size_t)u * USZ + r));

        asm volatile("s_wait_asynccnt 0" ::: "memory");   // LDS data landed

#pragma unroll
        for (int k = 0; k < NUPD; ++k)                    // lane reads only its own
            sum += *(const f32x4*)&sbuf[k * 1024 + t * 4];

        acc += 0.125f * sum;
        __builtin_nontemporal_store(acc,
            (f32x4*)(out + (size_t)c * OUT_LEN + (size_t)u * USZ + r));
        return;
    }

    // ---- edge rows (u < 7 or u > 255): generic bounds ----
    for (int k = k_lo; k <= k_hi; ++k)
        async_ld_b128(lds0 + (unsigned)(k * 4096), gbase + (long long)k * kstep);

    f32x4 acc = __builtin_nontemporal_load(
        (const f32x4*)(snapshot + (size_t)c * SNAP + (size_t)u * USZ + r));

    asm volatile("s_wait_asynccnt 0" ::: "memory");

    for (int k = k_lo; k <= k_hi; ++k)
        sum += *(const f32x4*)&sbuf[k * 1024 + t * 4];
    acc += 0.125f * sum;

    float* dst;
    if (u < BATCH)   // leading edge rows still belong to `output`
        dst = out + (size_t)c * OUT_LEN + (size_t)u * USZ + r;
    else             // trailing rows are the live head of new_snapshot
        dst = out + (size_t)NCH * OUT_LEN + (size_t)c * SNAP
                  + (size_t)(u - BATCH) * USZ + r;
    __builtin_nontemporal_store(acc, (f32x4*)dst);
}

// Zero-fill new_snapshot[:, REST:SNAP)  (harness poisons d_out).
__global__ __launch_bounds__(256) void oa_zero(float* __restrict__ out)
{
    const size_t i4      = (size_t)blockIdx.x * blockDim.x + threadIdx.x;
    const size_t per_ch4 = (size_t)(SNAP - REST) / 4;         // 262144 float4 / ch
    const int    c       = (int)(i4 / per_ch4);
    const size_t j       = (i4 % per_ch4) * 4;
    f32x4 z = (f32x4)0.f;
    __builtin_nontemporal_store(z,
        (f32x4*)(out + (size_t)NCH * OUT_LEN + (size_t)c * SNAP + REST + j));
}

extern "C" void kernel_launch(void* const* d_in, const int* in_sizes, int n_in,
                              void* d_out, int out_size, void* d_ws, size_t ws_size,
                              hipStream_t stream)
{
    const float* update   = (const float*)d_in[0];   // (256, 2, 32768) f32
    const float* snapshot = (const float*)d_in[1];   // (2, 1077248)    f32
    float*       out      = (float*)d_out;           // 2*OUT_LEN + 2*SNAP f32

    dim3 grid(4, NU, NCH);                           // tile, u-row, channel
    oa_main<<<grid, 256, 0, stream>>>(update, snapshot, out);

    // 2 * (SNAP - REST) / 4 = 524288 float4 -> 2048 blocks of 256 threads.
    oa_zero<<<2048, 256, 0, stream>>>(out);
}